// SequenceSummarizer_11046655885765
// MI455X (gfx1250) — compile-verified
//
#include <hip/hip_runtime.h>

typedef __attribute__((ext_vector_type(2))) float v2f;
typedef __attribute__((ext_vector_type(4))) float v4f;
typedef __attribute__((ext_vector_type(8))) float v8f;

#define BQ 8
#define SQ 4096
#define DQ 1024
#define LQ 16

// workspace layout (in floats)
#define WS_P    0                        // scores / probs : B*L*S = 524288 floats (2 MB)
#define WS_Q    (BQ * LQ * SQ)           // q[b,s]         : B*S   = 32768 floats
#define WS_PART (WS_Q + BQ * SQ)         // partial[b][16][D] = 131072 floats

// ---------------------------------------------------------------------------
// Kernel 1: scores[b,l,s] = (summarizer @ memory^T) / sqrt(D) via f32 WMMA.
// One wave per 16(L) x 16(S) tile; 256 k-steps of V_WMMA_F32_16X16X4_F32.
// A (16x4): lanes 0-15 = M rows, VGPR0/1 = {K0,K1} (lo half) / {K2,K3} (hi half)
// B (4x16): lanes 0-15 = N cols, same K split -> both fragments are float2
//           loads at offset k0 + 2*(lane>>4), 8-byte aligned.
// ---------------------------------------------------------------------------
__global__ __launch_bounds__(256) void k_scores(const float* __restrict__ mem,
                                                const float* __restrict__ summ,
                                                float* __restrict__ p) {
  const int lane = threadIdx.x & 31;
  const int wave = threadIdx.x >> 5;
  const int tile = blockIdx.x * 8 + wave;      // 2048 tiles = B * S/16
  const int b    = tile >> 8;                  // tile / (S/16)
  const int s0   = (tile & 255) << 4;
  const int n    = lane & 15;                  // A: row M, B: col N, C: col N
  const int hi   = lane >> 4;
  const int ko   = hi * 2;

  const float* Arow = summ + (size_t)n * DQ;                       // summarizer row l=n
  const float* Brow = mem + ((size_t)b * SQ + s0 + n) * DQ;        // memory row s=s0+n

  v8f c = {0.f, 0.f, 0.f, 0.f, 0.f, 0.f, 0.f, 0.f};
#pragma unroll 8
  for (int k0 = 0; k0 < DQ; k0 += 4) {
    v2f a  = *(const v2f*)(Arow + k0 + ko);
    v2f bb = *(const v2f*)(Brow + k0 + ko);
    c = __builtin_amdgcn_wmma_f32_16x16x4_f32(
        /*neg_a=*/false, a, /*neg_b=*/false, bb,
        /*c_mod=*/(short)0, c, /*reuse_a=*/false, /*reuse_b=*/false);
  }

  const float scale = 0.03125f;  // 1/sqrt(1024)
  // C layout: VGPR i holds M=i (lanes 0-15) and M=8+i (lanes 16-31), N=lane&15
  float* prow = p + ((size_t)(b * LQ + hi * 8)) * SQ + (s0 + n);
#pragma unroll
  for (int i = 0; i < 8; ++i)
    prow[(size_t)i * SQ] = c[i] * scale;
}

// ---------------------------------------------------------------------------
// Kernel 2: softmax over S for each of the B*L = 128 rows (in place).
// ---------------------------------------------------------------------------
__global__ __launch_bounds__(256) void k_softmax(float* __restrict__ p) {
  __shared__ float red[256];
  float* pr   = p + (size_t)blockIdx.x * SQ;
  const int t = threadIdx.x;

  float vals[16];
  float m = -1e30f;
#pragma unroll
  for (int i = 0; i < 16; ++i) {
    vals[i] = pr[t + i * 256];
    m = fmaxf(m, vals[i]);
  }
  red[t] = m;
  __syncthreads();
  for (int off = 128; off > 0; off >>= 1) {
    if (t < off) red[t] = fmaxf(red[t], red[t + off]);
    __syncthreads();
  }
  m = red[0];
  __syncthreads();

  float s = 0.f;
#pragma unroll
  for (int i = 0; i < 16; ++i) {
    vals[i] = __expf(vals[i] - m);
    s += vals[i];
  }
  red[t] = s;
  __syncthreads();
  for (int off = 128; off > 0; off >>= 1) {
    if (t < off) red[t] += red[t + off];
    __syncthreads();
  }
  const float inv = 1.0f / red[0];
#pragma unroll
  for (int i = 0; i < 16; ++i) pr[t + i * 256] = vals[i] * inv;
}

// ---------------------------------------------------------------------------
// Kernel 3: fold projection: q[b,s] = sum_l w[l] * p[b,l,s]
// ---------------------------------------------------------------------------
__global__ __launch_bounds__(256) void k_q(const float* __restrict__ p,
                                           const float* __restrict__ w,
                                           float* __restrict__ q) {
  const int idx = blockIdx.x * 256 + threadIdx.x;   // 0 .. B*S-1
  const int b   = idx >> 12;                        // / 4096
  const int s   = idx & 4095;
  const float* pb = p + ((size_t)b * LQ) * SQ + s;
  float acc = 0.f;
#pragma unroll
  for (int l = 0; l < LQ; ++l) acc += w[l] * pb[(size_t)l * SQ];
  q[idx] = acc;
}

// ---------------------------------------------------------------------------
// Kernel 4: partial[b][chunk][d] = sum_{s in 256-chunk} q[b,s] * mem[b,s,d]
// Second streaming pass over memory (should hit L2: 128MB < 192MB).
// ---------------------------------------------------------------------------
__global__ __launch_bounds__(256) void k_ctx(const float* __restrict__ mem,
                                             const float* __restrict__ q,
                                             float* __restrict__ part) {
  const int b  = blockIdx.x >> 4;
  const int ch = blockIdx.x & 15;
  const int s0 = ch * 256;
  const int d  = threadIdx.x * 4;

  const float* mb = mem + ((size_t)b * SQ + s0) * DQ + d;
  const float* qb = q + b * SQ + s0;

  v4f acc = {0.f, 0.f, 0.f, 0.f};
  for (int s = 0; s < 256; ++s) {
    const float qv = qb[s];
    v4f mvec = *(const v4f*)(mb + (size_t)s * DQ);
    acc += qv * mvec;
  }
  *(v4f*)(part + (size_t)blockIdx.x * DQ + d) = acc;
}

// ---------------------------------------------------------------------------
// Kernel 5: out[b,d] = sum over 16 chunks of partial
// ---------------------------------------------------------------------------
__global__ __launch_bounds__(256) void k_out(const float* __restrict__ part,
                                             float* __restrict__ out) {
  const int idx = blockIdx.x * 256 + threadIdx.x;   // 0 .. 8191
  const int b   = idx >> 10;
  const int d   = idx & 1023;
  float acc = 0.f;
#pragma unroll
  for (int c = 0; c < 16; ++c)
    acc += part[((size_t)(b * 16 + c)) * DQ + d];
  out[idx] = acc;
}

extern "C" void kernel_launch(void* const* d_in, const int* in_sizes, int n_in,
                              void* d_out, int out_size, void* d_ws, size_t ws_size,
                              hipStream_t stream) {
  const float* mem  = (const float*)d_in[0];   // [B,S,D] f32
  const float* summ = (const float*)d_in[1];   // [L,D]   f32
  const float* w    = (const float*)d_in[2];   // [1,L]   f32
  float* out = (float*)d_out;                  // [B,1,D] f32
  float* ws  = (float*)d_ws;

  float* p    = ws + WS_P;
  float* q    = ws + WS_Q;
  float* part = ws + WS_PART;

  k_scores <<<256,           256, 0, stream>>>(mem, summ, p);
  k_softmax<<<BQ * LQ,       256, 0, stream>>>(p);
  k_q      <<<(BQ * SQ)/256, 256, 0, stream>>>(p, w, q);
  k_ctx    <<<BQ * 16,       256, 0, stream>>>(mem, q, part);
  k_out    <<<(BQ * DQ)/256, 256, 0, stream>>>(part, out);
}